// Model_81183471829005
// MI455X (gfx1250) — compile-verified
//
#include <hip/hip_runtime.h>
#include <math.h>

typedef __attribute__((ext_vector_type(2))) float v2f;
typedef __attribute__((ext_vector_type(4))) float f4;
typedef __attribute__((ext_vector_type(8))) float v8f;

constexpr int NNODES = 512;
constexpr int NEDGES = 50;
constexpr int NE     = 1600;
constexpr int BB     = 64;
constexpr int CC     = 512;
constexpr float NEG_SLOPE = 0.2f;
constexpr float MARGIN    = 4.2f;

// ---------------------------------------------------------------------------
// 1) GEMM: xw[n,b,c] = sum_k x[b,n,k] * W[k,c]   via V_WMMA_F32_16X16X4_F32
//    GEMM rows = b*512 + n (flat over x), stored permuted to [N, B, C].
//    128x128 tile / workgroup, 8 waves, each wave 32x64 (2x4 16x16 subtiles).
//    LDS is K-pair interleaved (v2f) so each WMMA fragment is ONE ds_load_b64
//    into an aligned VGPR pair (no repacking moves). Double-buffered, with
//    global loads for chunk i+1 issued before the WMMA burst of chunk i.
// ---------------------------------------------------------------------------
#define KC 32               // K per LDS chunk
#define KP (KC / 2)         // float2 K-pairs per chunk
#define PITCH 132           // float2 pitch per K-pair row

__global__ __launch_bounds__(256) void k_gemm(const float* __restrict__ x,
                                              const float* __restrict__ w,
                                              float* __restrict__ xw) {
  // As2[buf][kp][row] = { A[2kp][row], A[2kp+1][row] }
  // Bs2[buf][kp][col] = { B[2kp][col], B[2kp+1][col] }
  __shared__ __align__(16) v2f As2[2][KP][PITCH];
  __shared__ __align__(16) v2f Bs2[2][KP][PITCH];

  const int tid  = threadIdx.x;
  const int lane = tid & 31;
  const int wave = tid >> 5;
  const int wr = wave & 3;            // wave row group (32 rows each)
  const int wc = wave >> 2;           // wave col group (64 cols each)
  const int rowBlock = blockIdx.y * 128;
  const int colBlock = blockIdx.x * 128;
  const int rb = wr * 32, cb = wc * 64;
  const int lhalf = lane >> 4;        // 0: K pair {2k,2k+1}, 1: K pair {2k+2,2k+3}
  const int l15   = lane & 15;

  v8f acc[2][4];
#pragma unroll
  for (int i = 0; i < 2; ++i)
#pragma unroll
    for (int j = 0; j < 4; ++j) acc[i][j] = (v8f)0.0f;

  // staging registers
  f4 ra[4];           // A: 4 float4 (4 k's of one row each)
  f4 rb0[2], rb1[2];  // B: rows 2kp and 2kp+1, 4 cols each

  auto load_global = [&](int k0) {
#pragma unroll
    for (int i = 0; i < 4; ++i) {
      int q = tid + i * 256;                  // 0..1023
      int row = q >> 3;                       // 128 rows x 8 float4
      int kq  = (q & 7) * 4;
      ra[i] = *(const f4*)(x + (size_t)(rowBlock + row) * CC + k0 + kq);
    }
#pragma unroll
    for (int j = 0; j < 2; ++j) {
      int q  = tid + j * 256;                 // 0..511
      int kp = q >> 5;                        // 16 k-pairs x 32 col-quads
      int cg = (q & 31) * 4;
      rb0[j] = *(const f4*)(w + (size_t)(k0 + 2 * kp + 0) * CC + colBlock + cg);
      rb1[j] = *(const f4*)(w + (size_t)(k0 + 2 * kp + 1) * CC + colBlock + cg);
    }
  };

  auto store_lds = [&](int buf) {
#pragma unroll
    for (int i = 0; i < 4; ++i) {
      int q = tid + i * 256;
      int row = q >> 3;
      int kq  = (q & 7) * 4;
      v2f lo; lo.x = ra[i].x; lo.y = ra[i].y;
      v2f hi; hi.x = ra[i].z; hi.y = ra[i].w;
      As2[buf][(kq >> 1) + 0][row] = lo;      // K = {kq, kq+1}
      As2[buf][(kq >> 1) + 1][row] = hi;      // K = {kq+2, kq+3}
    }
#pragma unroll
    for (int j = 0; j < 2; ++j) {
      int q  = tid + j * 256;
      int kp = q >> 5;
      int cg = (q & 31) * 4;
      f4 p0; p0.x = rb0[j].x; p0.y = rb1[j].x; p0.z = rb0[j].y; p0.w = rb1[j].y;
      f4 p1; p1.x = rb0[j].z; p1.y = rb1[j].z; p1.z = rb0[j].w; p1.w = rb1[j].w;
      *(f4*)&Bs2[buf][kp][cg + 0] = p0;       // cols cg, cg+1 interleaved pairs
      *(f4*)&Bs2[buf][kp][cg + 2] = p1;       // cols cg+2, cg+3
    }
  };

  constexpr int NCHUNK = CC / KC;             // 16
  load_global(0);
  store_lds(0);

  for (int ch = 0; ch < NCHUNK; ++ch) {
    __syncthreads();
    const int buf = ch & 1;
    if (ch + 1 < NCHUNK) load_global((ch + 1) * KC);   // issue early, hide latency

#pragma unroll
    for (int ks = 0; ks < KC; ks += 4) {
      const int kp = (ks >> 1) + lhalf;       // lanes 0-15: K={ks,ks+1}; 16-31: K={ks+2,ks+3}
      v2f a[2], b[4];
#pragma unroll
      for (int tr = 0; tr < 2; ++tr) a[tr] = As2[buf][kp][rb + tr * 16 + l15];
#pragma unroll
      for (int tc = 0; tc < 4; ++tc) b[tc] = Bs2[buf][kp][cb + tc * 16 + l15];
#pragma unroll
      for (int tr = 0; tr < 2; ++tr)
#pragma unroll
        for (int tc = 0; tc < 4; ++tc)
          acc[tr][tc] = __builtin_amdgcn_wmma_f32_16x16x4_f32(
              false, a[tr], false, b[tc], (short)0, acc[tr][tc], false, false);
    }

    if (ch + 1 < NCHUNK) store_lds(buf ^ 1);  // other buffer; barrier at loop top
  }

  // ---- epilogue: C/D layout: VGPR r -> M = r + 8*lane_half, N = lane&15.
  // Store permuted: GEMM row = b*512 + n  ==>  xw[(n*64 + b)*512 + col]
#pragma unroll
  for (int tr = 0; tr < 2; ++tr)
#pragma unroll
    for (int tc = 0; tc < 4; ++tc)
#pragma unroll
      for (int r = 0; r < 8; ++r) {
        int grow = rowBlock + rb + tr * 16 + r + lhalf * 8;
        int gcol = colBlock + cb + tc * 16 + l15;
        int bidx = grow >> 9;
        int n    = grow & 511;
        xw[((size_t)n * BB + bidx) * CC + gcol] = acc[tr][tc][r];
      }
}

// ---------------------------------------------------------------------------
// degree counting (integer atomics: deterministic)
// ---------------------------------------------------------------------------
__global__ void k_zero_counts(int* nodeCnt, int* edgeCnt) {
  int t = blockIdx.x * 256 + threadIdx.x;
  if (t < NNODES) nodeCnt[t] = 0;
  if (t < NEDGES) edgeCnt[t] = 0;
}

__global__ void k_degrees(const int* __restrict__ node_idx,
                          const int* __restrict__ edge_idx,
                          int* nodeCnt, int* edgeCnt) {
  int e = blockIdx.x * 256 + threadIdx.x;
  if (e < NE) {
    atomicAdd(&nodeCnt[node_idx[e]], 1);
    atomicAdd(&edgeCnt[edge_idx[e]], 1);
  }
}

// ---------------------------------------------------------------------------
// edge_sums[m,b,c] = sum over incidences of edge m of xw[node,b,c]
// ---------------------------------------------------------------------------
__global__ __launch_bounds__(256) void k_edge_sums(const float* __restrict__ xw,
                                                   const int* __restrict__ node_idx,
                                                   const int* __restrict__ edge_idx,
                                                   float* __restrict__ es) {
  const int m = blockIdx.y;
  const int off = blockIdx.x * 1024 + threadIdx.x * 4;  // within B*C=32768
  f4 acc = (f4)0.0f;
  for (int e = 0; e < NE; ++e) {
    if (edge_idx[e] == m) {
      int n = node_idx[e];
      f4 v = *(const f4*)(xw + (size_t)n * (BB * CC) + off);
      acc += v;
    }
  }
  *(f4*)(es + (size_t)m * (BB * CC) + off) = acc;
}

// ---------------------------------------------------------------------------
// per-edge: ||es[m,b,:]||^2 and <es[m,b,:], att[C:2C]>
// ---------------------------------------------------------------------------
__global__ __launch_bounds__(64) void k_edge_stats(const float* __restrict__ es,
                                                   const float* __restrict__ att,
                                                   float* __restrict__ norms2,
                                                   float* __restrict__ edge_att) {
  int m = blockIdx.x, b = threadIdx.x;
  const float* p  = es + ((size_t)m * BB + b) * CC;
  const float* a2 = att + CC;
  float s2 = 0.f, sa = 0.f;
  for (int c = 0; c < CC; ++c) { float v = p[c]; s2 = fmaf(v, v, s2); sa = fmaf(v, a2[c], sa); }
  norms2[m * BB + b]   = s2;
  edge_att[m * BB + b] = sa;
}

// per-node: <xw[n,b,:], att[0:C]>
__global__ __launch_bounds__(64) void k_node_att(const float* __restrict__ xw,
                                                 const float* __restrict__ att,
                                                 float* __restrict__ node_att) {
  int n = blockIdx.x, b = threadIdx.x;
  const float* p = xw + ((size_t)n * BB + b) * CC;
  float s = 0.f;
  for (int c = 0; c < CC; ++c) s = fmaf(p[c], att[c], s);
  node_att[n * BB + b] = s;
}

// ---------------------------------------------------------------------------
// alpha[e,b]: leaky-relu logit, segment-softmax over incidences of each node.
// Node-centric scan of index list: deterministic, no float atomics.
// ---------------------------------------------------------------------------
__global__ __launch_bounds__(64) void k_softmax(const float* __restrict__ node_att,
                                                const float* __restrict__ edge_att,
                                                const int* __restrict__ node_idx,
                                                const int* __restrict__ edge_idx,
                                                float* __restrict__ alpha) {
  int n = blockIdx.x, b = threadIdx.x;
  float na = node_att[n * BB + b];
  float mx = -3.4e38f;
  for (int e = 0; e < NE; ++e)
    if (node_idx[e] == n) {
      float v = na + edge_att[edge_idx[e] * BB + b];
      v = v > 0.f ? v : NEG_SLOPE * v;
      mx = fmaxf(mx, v);
    }
  float den = 0.f;
  for (int e = 0; e < NE; ++e)
    if (node_idx[e] == n) {
      float v = na + edge_att[edge_idx[e] * BB + b];
      v = v > 0.f ? v : NEG_SLOPE * v;
      den += __expf(v - mx);
    }
  float inv = den > 0.f ? 1.f / den : 0.f;
  for (int e = 0; e < NE; ++e)
    if (node_idx[e] == n) {
      float v = na + edge_att[edge_idx[e] * BB + b];
      v = v > 0.f ? v : NEG_SLOPE * v;
      alpha[e * BB + b] = __expf(v - mx) * inv;
    }
}

// ---------------------------------------------------------------------------
// pairwise hyperedge loss from Gram identities:
//   dist^2 = ni2 + nj2 - 2*inner,  cos = inner/(sqrt(ni2)sqrt(nj2)+1e-8)
// ---------------------------------------------------------------------------
__global__ __launch_bounds__(64) void k_loss_pairs(const float* __restrict__ es,
                                                   const float* __restrict__ norms2,
                                                   float* __restrict__ lossPart) {
  int pair = blockIdx.x;
  int i = pair / NEDGES, j = pair % NEDGES;
  int b = threadIdx.x;
  const float* pi = es + ((size_t)i * BB + b) * CC;
  const float* pj = es + ((size_t)j * BB + b) * CC;
  float inner = 0.f;
  for (int c = 0; c < CC; ++c) inner = fmaf(pi[c], pj[c], inner);
  float ni2 = norms2[i * BB + b], nj2 = norms2[j * BB + b];
  float cosa = inner / (sqrtf(ni2) * sqrtf(nj2) + 1e-8f);
  float dist = sqrtf(fmaxf(ni2 + nj2 - 2.f * inner, 0.f));
  float item = cosa * dist + (1.f - cosa) * fmaxf(MARGIN - dist, 0.f);

  __shared__ float red[64];
  red[b] = item;
  __syncthreads();
  for (int s = 32; s > 0; s >>= 1) {
    if (b < s) red[b] += red[b + s];
    __syncthreads();
  }
  if (b == 0) lossPart[pair] = red[0];
}

// ---------------------------------------------------------------------------
// propagate 1: out1[m,b,c] = (1/Bdeg[m]) * sum_{e in edge m} alpha[e,b]*xw[n_e,b,c]
// ---------------------------------------------------------------------------
__global__ __launch_bounds__(256) void k_out1(const float* __restrict__ xw,
                                              const float* __restrict__ alpha,
                                              const int* __restrict__ node_idx,
                                              const int* __restrict__ edge_idx,
                                              const int* __restrict__ edgeCnt,
                                              float* __restrict__ out1) {
  const int m = blockIdx.y;
  const int off = blockIdx.x * 1024 + threadIdx.x * 4;
  const int b = off >> 9;
  f4 acc = (f4)0.0f;
  for (int e = 0; e < NE; ++e) {
    if (edge_idx[e] == m) {
      float w = alpha[e * BB + b];
      f4 v = *(const f4*)(xw + (size_t)node_idx[e] * (BB * CC) + off);
      acc += w * v;
    }
  }
  int cnt = edgeCnt[m];
  float bn = cnt > 0 ? 1.f / (float)cnt : 0.f;
  acc *= bn;
  *(f4*)(out1 + (size_t)m * (BB * CC) + off) = acc;
}

// ---------------------------------------------------------------------------
// propagate 2: out[n,b,c] = D[n] * sum_{e: node_e==n} alpha[e,b]*out1[m_e,b,c]
// ---------------------------------------------------------------------------
__global__ __launch_bounds__(256) void k_out(const float* __restrict__ out1,
                                             const float* __restrict__ alpha,
                                             const int* __restrict__ node_idx,
                                             const int* __restrict__ edge_idx,
                                             const int* __restrict__ nodeCnt,
                                             float* __restrict__ out) {
  const int n = blockIdx.y;
  const int off = blockIdx.x * 1024 + threadIdx.x * 4;
  const int b = off >> 9;
  f4 acc = (f4)0.0f;
  for (int e = 0; e < NE; ++e) {
    if (node_idx[e] == n) {
      float w = alpha[e * BB + b];
      f4 v = *(const f4*)(out1 + (size_t)edge_idx[e] * (BB * CC) + off);
      acc += w * v;
    }
  }
  float D = (float)nodeCnt[n];
  acc *= D;
  *(f4*)(out + (size_t)n * (BB * CC) + off) = acc;
}

// ---------------------------------------------------------------------------
// per-node / per-edge total sums (for mean(x_i - x_j))
// ---------------------------------------------------------------------------
__global__ __launch_bounds__(256) void k_sums(const float* __restrict__ xw,
                                              const float* __restrict__ es,
                                              float* __restrict__ S_node,
                                              float* __restrict__ S_edge) {
  int i = blockIdx.x;
  const float* p = (i < NNODES) ? xw + (size_t)i * (BB * CC)
                                : es + (size_t)(i - NNODES) * (BB * CC);
  float s = 0.f;
  for (int t = threadIdx.x; t < BB * CC; t += 256) s += p[t];
  __shared__ float red[256];
  red[threadIdx.x] = s;
  __syncthreads();
  for (int st = 128; st > 0; st >>= 1) {
    if (threadIdx.x < st) red[threadIdx.x] += red[threadIdx.x + st];
    __syncthreads();
  }
  if (threadIdx.x == 0) {
    if (i < NNODES) S_node[i] = red[0];
    else            S_edge[i - NNODES] = red[0];
  }
}

// ---------------------------------------------------------------------------
// final scalar: |mean(x_i - x_j)| + |mean(loss_items)| / (M+1)^2
// ---------------------------------------------------------------------------
__global__ __launch_bounds__(256) void k_final(const float* __restrict__ lossPart,
                                               const float* __restrict__ S_node,
                                               const float* __restrict__ S_edge,
                                               const int* __restrict__ node_idx,
                                               const int* __restrict__ edge_idx,
                                               float* __restrict__ loss_out) {
  __shared__ float red[256];
  int tid = threadIdx.x;

  float lh = 0.f;
  for (int p = tid; p < NEDGES * NEDGES; p += 256) lh += lossPart[p];
  red[tid] = lh;
  __syncthreads();
  for (int s = 128; s > 0; s >>= 1) {
    if (tid < s) red[tid] += red[tid + s];
    __syncthreads();
  }
  float lhTot = red[0];
  __syncthreads();

  float df = 0.f;
  for (int e = tid; e < NE; e += 256) df += S_node[node_idx[e]] - S_edge[edge_idx[e]];
  red[tid] = df;
  __syncthreads();
  for (int s = 128; s > 0; s >>= 1) {
    if (tid < s) red[tid] += red[tid + s];
    __syncthreads();
  }
  if (tid == 0) {
    float loss_hyper = fabsf(lhTot / (float)(NEDGES * NEDGES * BB)) /
                       (float)((NEDGES + 1) * (NEDGES + 1));
    float term1 = fabsf(red[0] / ((float)NE * (float)BB * (float)CC));
    loss_out[0] = term1 + loss_hyper;
  }
}

// ---------------------------------------------------------------------------
extern "C" void kernel_launch(void* const* d_in, const int* in_sizes, int n_in,
                              void* d_out, int out_size, void* d_ws, size_t ws_size,
                              hipStream_t stream) {
  const float* x      = (const float*)d_in[0];   // [B, N, C]
  const float* weight = (const float*)d_in[1];   // [C, C]
  const float* att    = (const float*)d_in[2];   // [2C]
  const int* node_idx = (const int*)d_in[3];     // [E]
  const int* edge_idx = (const int*)d_in[4];     // [E]

  float* out = (float*)d_out;                         // [N, B, C]
  float* loss_out = out + (size_t)NNODES * BB * CC;   // scalar

  // workspace carve-up (256B aligned chunks)
  char* p = (char*)d_ws;
  auto take = [&](size_t bytes) {
    char* r = p;
    p += (bytes + 255) & ~(size_t)255;
    return r;
  };
  float* xw       = (float*)take((size_t)NNODES * BB * CC * 4);  // 67 MB
  float* es       = (float*)take((size_t)NEDGES * BB * CC * 4);  // 6.6 MB
  float* out1     = (float*)take((size_t)NEDGES * BB * CC * 4);  // 6.6 MB
  float* alpha    = (float*)take((size_t)NE * BB * 4);
  float* node_att = (float*)take((size_t)NNODES * BB * 4);
  float* edge_att = (float*)take((size_t)NEDGES * BB * 4);
  float* norms2   = (float*)take((size_t)NEDGES * BB * 4);
  float* lossPart = (float*)take((size_t)NEDGES * NEDGES * 4);
  float* S_node   = (float*)take((size_t)NNODES * 4);
  float* S_edge   = (float*)take((size_t)NEDGES * 4);
  int*   nodeCnt  = (int*)take((size_t)NNODES * 4);
  int*   edgeCnt  = (int*)take((size_t)NEDGES * 4);

  k_zero_counts<<<2, 256, 0, stream>>>(nodeCnt, edgeCnt);
  k_degrees<<<(NE + 255) / 256, 256, 0, stream>>>(node_idx, edge_idx, nodeCnt, edgeCnt);

  // main WMMA GEMM: rows = B*N = 32768 (256 row-tiles), cols = 512 (4 col-tiles)
  k_gemm<<<dim3(4, 256), 256, 0, stream>>>(x, weight, xw);

  k_edge_sums<<<dim3(32, NEDGES), 256, 0, stream>>>(xw, node_idx, edge_idx, es);
  k_edge_stats<<<NEDGES, 64, 0, stream>>>(es, att, norms2, edge_att);
  k_node_att<<<NNODES, 64, 0, stream>>>(xw, att, node_att);
  k_softmax<<<NNODES, 64, 0, stream>>>(node_att, edge_att, node_idx, edge_idx, alpha);
  k_loss_pairs<<<NEDGES * NEDGES, 64, 0, stream>>>(es, norms2, lossPart);
  k_out1<<<dim3(32, NEDGES), 256, 0, stream>>>(xw, alpha, node_idx, edge_idx, edgeCnt, out1);
  k_out<<<dim3(32, NNODES), 256, 0, stream>>>(out1, alpha, node_idx, edge_idx, nodeCnt, out);
  k_sums<<<NNODES + NEDGES, 256, 0, stream>>>(xw, es, S_node, S_edge);
  k_final<<<1, 256, 0, stream>>>(lossPart, S_node, S_edge, node_idx, edge_idx, loss_out);
}